// SimPLREncoder_50551765074246
// MI455X (gfx1250) — compile-verified
//
#include <hip/hip_runtime.h>
#include <hip/hip_bf16.h>

#define B_   2
#define L_   20000
#define D_   256
#define NS_  4
#define NC_  91
#define NQ_  300
#define NROW (B_ * L_)          // 40000
#define NTILES 24               // 384 padded cols / 16
#define ASTRIDE 260             // LDS row stride (floats) to dodge bank conflicts

typedef __attribute__((ext_vector_type(2))) float v2f;
typedef __attribute__((ext_vector_type(8))) float v8f;

// ---------------------------------------------------------------------------
// Kernel 0: pack w_cls [256 x 364] into padded fragment-friendly layout
// Bpad[nt][k][n] = w_cls[k][col(nt*16+n)], zeros in 91..95 padding lanes.
// ---------------------------------------------------------------------------
__global__ __launch_bounds__(256) void pack_b_kernel(const float* __restrict__ w_cls,
                                                     float* __restrict__ Bpad) {
    int i = blockIdx.x * 256 + threadIdx.x;      // over 24*256*16 = 98304
    if (i < NTILES * D_ * 16) {
        int n  = i & 15;
        int k  = (i >> 4) & 255;
        int nt = i >> 12;
        int colp = nt * 16 + n;
        int s = colp / 96;
        int j = colp - s * 96;
        float v = (j < NC_) ? w_cls[(size_t)k * (NS_ * NC_) + s * NC_ + j] : 0.0f;
        Bpad[i] = v;
    }
}

// ---------------------------------------------------------------------------
// Kernel 1: output = where(mask, 0, src)   (40 MB masked copy, float4)
// ---------------------------------------------------------------------------
__global__ __launch_bounds__(256) void copy_kernel(const float4* __restrict__ src,
                                                   const unsigned char* __restrict__ mask,
                                                   float4* __restrict__ out, int n4) {
    int i = blockIdx.x * 256 + threadIdx.x;
    if (i < n4) {
        float4 v = src[i];
        int row = i >> 6;               // 64 float4 per 256-float row
        if (mask[row]) { v.x = 0.f; v.y = 0.f; v.z = 0.f; v.w = 0.f; }
        out[i] = v;
    }
}

// ---------------------------------------------------------------------------
// Kernel 2: scores[gr*4+s] = softmax(src_row @ w_cls + b_cls)[class 0]
// fp32 WMMA 16x16x4. One block = 16 rows, 4 waves; each wave runs 3 passes of
// two column tiles sharing one A-fragment stream (dual accumulators).
// ---------------------------------------------------------------------------
__device__ __forceinline__ void score_epilogue(const v8f& c, int nt, int half,
                                               int lanelow, int wave,
                                               const float* __restrict__ b_cls,
                                               float (*wsum)[16][NS_],
                                               float (*cls0)[NS_]) {
    int colp = nt * 16 + lanelow;
    int s = colp / 96;
    int j = colp - s * 96;
    float vf = (j < NC_) ? 1.0f : 0.0f;
    int col = s * NC_ + (j < NC_ ? j : NC_ - 1);
    float bias = b_cls[col] * vf;
#pragma unroll
    for (int r = 0; r < 8; ++r) {
        int row = r + 8 * half;                  // C layout: vgpr r, half -> row
        float e = vf * __expf(c[r] + bias);
        if (j == 0) cls0[row][s] = e;            // class-0 numerator
        float es = e;
        es += __shfl_xor(es, 1, 16);
        es += __shfl_xor(es, 2, 16);
        es += __shfl_xor(es, 4, 16);
        es += __shfl_xor(es, 8, 16);
        if (lanelow == 0) wsum[wave][row][s] += es;  // single-writer, in order
    }
}

__global__ __launch_bounds__(128) void score_kernel(const float* __restrict__ src,
                                                    const unsigned char* __restrict__ mask,
                                                    const float* __restrict__ Bpad,
                                                    const float* __restrict__ b_cls,
                                                    float* __restrict__ scores) {
    __shared__ float Alds[16 * ASTRIDE];
    __shared__ float wsum[4][16][NS_];   // per-wave partial row sums (deterministic)
    __shared__ float cls0[16][NS_];

    const int tid  = threadIdx.x;
    const int lane = tid & 31;
    const int wave = tid >> 5;
    const int lanelow = lane & 15;
    const int half = lane >> 4;
    const int gr0 = blockIdx.x * 16;

    // zero accumulators
    for (int i = tid; i < 4 * 16 * NS_; i += 128) (&wsum[0][0][0])[i] = 0.f;
    for (int i = tid; i < 16 * NS_; i += 128) (&cls0[0][0])[i] = 0.f;

    // stage 16 rows (masked) into LDS: 1024 float4 transfers
    for (int i = tid; i < 1024; i += 128) {
        int r = i >> 6, cq = i & 63;
        float4 v = *(const float4*)(src + ((size_t)(gr0 + r)) * D_ + cq * 4);
        if (mask[gr0 + r]) { v.x = 0.f; v.y = 0.f; v.z = 0.f; v.w = 0.f; }
        *(float4*)(&Alds[r * ASTRIDE + cq * 4]) = v;
    }
    __syncthreads();

    // A fragment base: lane m holds row (lane&15), K pair offset 2*half
    const float* Arow = &Alds[lanelow * ASTRIDE + half * 2];

    for (int pp = 0; pp < 3; ++pp) {             // uniform per wave
        const int nt0 = wave + pp * 8;
        const int nt1 = nt0 + 4;
        const float* Bp0 = Bpad + (size_t)nt0 * (D_ * 16) + (half * 2) * 16 + lanelow;
        const float* Bp1 = Bpad + (size_t)nt1 * (D_ * 16) + (half * 2) * 16 + lanelow;

        v8f c0 = {0.f, 0.f, 0.f, 0.f, 0.f, 0.f, 0.f, 0.f};
        v8f c1 = {0.f, 0.f, 0.f, 0.f, 0.f, 0.f, 0.f, 0.f};
#pragma unroll 8
        for (int k = 0; k < D_; k += 4) {
            float2 a2 = *(const float2*)(Arow + k);
            v2f av = {a2.x, a2.y};
            v2f bv0 = {Bp0[k * 16], Bp0[k * 16 + 16]};
            v2f bv1 = {Bp1[k * 16], Bp1[k * 16 + 16]};
            c0 = __builtin_amdgcn_wmma_f32_16x16x4_f32(false, av, false, bv0,
                                                       (short)0, c0, false, false);
            c1 = __builtin_amdgcn_wmma_f32_16x16x4_f32(false, av, false, bv1,
                                                       (short)0, c1, false, false);
        }
        score_epilogue(c0, nt0, half, lanelow, wave, b_cls, wsum, cls0);
        score_epilogue(c1, nt1, half, lanelow, wave, b_cls, wsum, cls0);
    }
    __syncthreads();

    if (tid < 64) {
        int row = tid >> 2, sc = tid & 3;
        float ssum = wsum[0][row][sc] + wsum[1][row][sc] +
                     wsum[2][row][sc] + wsum[3][row][sc];
        int gr = gr0 + row;
        float v = mask[gr] ? -65504.0f : (cls0[row][sc] / ssum);
        scores[(size_t)gr * 4 + sc] = v;
    }
}

// ---------------------------------------------------------------------------
// Kernel 3: exact deterministic top-300 per batch (radix threshold + bitonic)
// ---------------------------------------------------------------------------
#define CAP 4096
__global__ __launch_bounds__(256) void topk_kernel(const float* __restrict__ scores,
                                                   int* __restrict__ topk) {
    const int b = blockIdx.x;
    const int t = threadIdx.x;
    const float* sc = scores + (size_t)b * (L_ * NS_);
    __shared__ unsigned int hist[1024];
    __shared__ unsigned long long keys[CAP];
    __shared__ int sh_T;
    __shared__ int cnt;

    for (int i = t; i < 1024; i += 256) hist[i] = 0u;
    if (t == 0) cnt = 0;
    __syncthreads();

    for (int i = t; i < L_ * NS_; i += 256) {
        unsigned int ub = __float_as_uint(sc[i]);
        unsigned int u = (ub & 0x80000000u) ? ~ub : (ub | 0x80000000u);
        atomicAdd(&hist[u >> 22], 1u);
    }
    __syncthreads();
    if (t == 0) {
        int cum = 0, T = 0;
        for (int bin = 1023; bin >= 0; --bin) {
            cum += (int)hist[bin];
            if (cum >= NQ_) { T = bin; break; }
        }
        sh_T = T;
    }
    for (int i = t; i < CAP; i += 256) keys[i] = 0xFFFFFFFFFFFFFFFFull; // pad = last
    __syncthreads();

    unsigned int T = (unsigned int)sh_T;
    for (int i = t; i < L_ * NS_; i += 256) {
        unsigned int ub = __float_as_uint(sc[i]);
        unsigned int u = (ub & 0x80000000u) ? ~ub : (ub | 0x80000000u);
        if ((u >> 22) >= T) {
            int p = atomicAdd(&cnt, 1);
            if (p < CAP) {
                unsigned long long comp =
                    ((unsigned long long)u << 32) | (unsigned int)(~(unsigned int)i);
                keys[p] = ~comp;   // ascending sort -> best score, lowest idx first
            }
        }
    }
    __syncthreads();

    for (int k = 2; k <= CAP; k <<= 1) {
        for (int j = k >> 1; j > 0; j >>= 1) {
            for (int i = t; i < CAP; i += 256) {
                int ij = i ^ j;
                if (ij > i) {
                    unsigned long long a = keys[i], bb = keys[ij];
                    bool up = ((i & k) == 0);
                    bool sw = up ? (a > bb) : (a < bb);
                    if (sw) { keys[i] = bb; keys[ij] = a; }
                }
            }
            __syncthreads();
        }
    }
    for (int i = t; i < NQ_; i += 256) {
        unsigned long long comp = ~keys[i];
        int idx = (int)(~(unsigned int)(comp & 0xFFFFFFFFull));
        topk[b * NQ_ + i] = idx;
    }
}

// ---------------------------------------------------------------------------
// Kernel 4: per selected query: bbox refine + sigmoid, pos embed, embed GEMV+LN
// ---------------------------------------------------------------------------
__global__ __launch_bounds__(256) void finalize_kernel(
    const float* __restrict__ src, const unsigned char* __restrict__ mask,
    const float* __restrict__ ref_windows,
    const float* __restrict__ w_bbox, const float* __restrict__ b_bbox,
    const float* __restrict__ w_enc, const float* __restrict__ b_enc,
    const float* __restrict__ ln_g, const float* __restrict__ ln_b,
    const int* __restrict__ topk,
    float* __restrict__ out_embed, float* __restrict__ out_ref,
    float* __restrict__ out_pos) {
    const int b = blockIdx.x / NQ_;
    const int q = blockIdx.x % NQ_;
    const int t = threadIdx.x;
    const int idx = topk[b * NQ_ + q];
    const int l = idx >> 2;
    const int s = idx & 3;

    __shared__ float ev[D_];
    __shared__ float red[D_];
    __shared__ float coords[4];
    __shared__ float ps[8], ps2[8];
    __shared__ float sh_mu, sh_rstd;

    bool m = mask[b * L_ + l] != 0;
    ev[t] = m ? 0.f : src[((size_t)b * L_ + l) * D_ + t];
    __syncthreads();

    // bbox: 4 coords, each computed by a 64-thread group
    {
        int g = t >> 6, lt = t & 63;
        float p = 0.f;
        for (int kk = lt; kk < D_; kk += 64)
            p += ev[kk] * w_bbox[kk * (NS_ * 4) + s * 4 + g];
        red[t] = p;
    }
    __syncthreads();
    if (t < 4) {
        float dot = b_bbox[s * 4 + t];
        for (int kk = 0; kk < 64; ++kk) dot += red[t * 64 + kk];
        float r = ref_windows[(((size_t)b * L_ + l) * NS_ + s) * 4 + t];
        r = fminf(fmaxf(r, 0.f), 1.f);
        float x1 = fmaxf(r, 1e-5f);
        float x2 = fmaxf(1.f - r, 1e-5f);
        float val = dot + __logf(x1 / x2);
        float sg = 1.f / (1.f + __expf(-val));
        coords[t] = sg;
        out_ref[((size_t)b * NQ_ + q) * 4 + t] = sg;
    }
    __syncthreads();

    // positional embedding: pos(xy) + size(wh)
    {
        int c = t >> 7;
        int mf = (t & 127) >> 1;
        int isc = t & 1;
        // freq = 10000^(mf/64) = 2^(mf * log2(10000)/64)
        float freq = exp2f((float)mf * 0.20762050595f);
        float scl = 6.28318530717958647692f / freq;
        float a1 = coords[c] * scl;
        float a2 = coords[c + 2] * scl;
        float v = isc ? (__cosf(a1) + __cosf(a2)) : (__sinf(a1) + __sinf(a2));
        out_pos[((size_t)b * NQ_ + q) * D_ + t] = v;
    }

    // embed GEMV + layernorm
    float acc = b_enc[t];
    for (int kk = 0; kk < D_; ++kk) acc += ev[kk] * w_enc[kk * D_ + t];
    {
        float v = acc, v2 = acc * acc;
        for (int mm = 16; mm >= 1; mm >>= 1) {
            v  += __shfl_xor(v,  mm, 32);
            v2 += __shfl_xor(v2, mm, 32);
        }
        int wid = t >> 5;
        if ((t & 31) == 0) { ps[wid] = v; ps2[wid] = v2; }
    }
    __syncthreads();
    if (t == 0) {
        float sA = 0.f, sB = 0.f;
        for (int i = 0; i < 8; ++i) { sA += ps[i]; sB += ps2[i]; }
        float mu = sA / (float)D_;
        float var = sB / (float)D_ - mu * mu;
        sh_mu = mu;
        sh_rstd = rsqrtf(var + 1e-5f);
    }
    __syncthreads();
    out_embed[((size_t)b * NQ_ + q) * D_ + t] =
        (acc - sh_mu) * sh_rstd * ln_g[t] + ln_b[t];
}

// ---------------------------------------------------------------------------
extern "C" void kernel_launch(void* const* d_in, const int* in_sizes, int n_in,
                              void* d_out, int out_size, void* d_ws, size_t ws_size,
                              hipStream_t stream) {
    const float* src        = (const float*)d_in[0];
    const unsigned char* sm = (const unsigned char*)d_in[1];
    const float* ref_w      = (const float*)d_in[2];
    const float* w_bbox     = (const float*)d_in[3];
    const float* b_bbox     = (const float*)d_in[4];
    const float* w_cls      = (const float*)d_in[5];
    const float* b_cls      = (const float*)d_in[6];
    const float* w_enc      = (const float*)d_in[7];
    const float* b_enc      = (const float*)d_in[8];
    const float* ln_g       = (const float*)d_in[9];
    const float* ln_b       = (const float*)d_in[10];

    float* out = (float*)d_out;
    const size_t n_output = (size_t)B_ * L_ * D_;        // 10,240,000
    const size_t n_embed  = (size_t)B_ * NQ_ * D_;       // 153,600
    const size_t n_ref    = (size_t)B_ * NQ_ * 4;        // 2,400
    float* out_output = out;
    float* out_embed  = out + n_output;
    float* out_ref    = out + n_output + n_embed;
    float* out_pos    = out + n_output + n_embed + n_ref;

    float* scores = (float*)d_ws;                                    // 160,000 f
    int*   topk   = (int*)((char*)d_ws + (size_t)B_ * L_ * NS_ * 4); // 600 ints
    float* Bpad   = (float*)((char*)d_ws + 786432);                  // 98,304 f

    // 0) pack padded B
    pack_b_kernel<<<dim3((NTILES * D_ * 16 + 255) / 256), dim3(256), 0, stream>>>(
        w_cls, Bpad);

    // 1) output copy
    int n4 = (int)(n_output / 4);
    copy_kernel<<<dim3((n4 + 255) / 256), dim3(256), 0, stream>>>(
        (const float4*)src, sm, (float4*)out_output, n4);

    // 2) objectness scores via fp32 WMMA
    score_kernel<<<dim3(NROW / 16), dim3(128), 0, stream>>>(
        src, sm, Bpad, b_cls, scores);

    // 3) top-300 per batch
    topk_kernel<<<dim3(B_), dim3(256), 0, stream>>>(scores, topk);

    // 4) per-query outputs
    finalize_kernel<<<dim3(B_ * NQ_), dim3(256), 0, stream>>>(
        src, sm, ref_w, w_bbox, b_bbox, w_enc, b_enc, ln_g, ln_b, topk,
        out_embed, out_ref, out_pos);
}